// MBNCEHead_56865366999091
// MI455X (gfx1250) — compile-verified
//
#include <hip/hip_runtime.h>

// ---------------------------------------------------------------------------
// MBNCE head for MI455X (gfx1250).
// HBM-bound (1.07 GB streamed, ~46us floor at 23.3 TB/s).
// Path: TDM tensor_load_to_lds (double-buffered, padded LDS rows) ->
//       ds_load_b64 in WMMA A-layout -> v_wmma_f32_16x16x4_f32 chain ->
//       exp + per-n atomics -> tiny finalize kernel for the 9 scalars.
// ---------------------------------------------------------------------------

#define N_BATCH 128
#define C_DIM   256
#define BANK1   4097          // bank_size + 1
#define T_INV   10.0f         // 1 / TEMPERATURE

#define ROWS_PER_BLOCK 64
#define WAVES   4
#define THREADS (WAVES * 32)
#define ROW_STRIDE_F 260                       // 256 + 4 pad floats (bank-conflict-free)
#define TILE_BYTES (ROWS_PER_BLOCK * ROW_STRIDE_F * 4)   // 66560 B
#define NBX     8
#define NTILES  ((BANK1 + ROWS_PER_BLOCK - 1) / ROWS_PER_BLOCK)  // 65

typedef float v2f __attribute__((ext_vector_type(2)));
typedef float v8f __attribute__((ext_vector_type(8)));
typedef unsigned int v4u __attribute__((ext_vector_type(4)));
typedef int v8i __attribute__((ext_vector_type(8)));
typedef int v4i __attribute__((ext_vector_type(4)));

// ---------------------------------------------------------------------------
// TDM: load a [rows=64, cols=256] f32 tile (row stride = 256 floats in HBM)
// into LDS with 16B padding after every 1024B row -> 1040B LDS row stride.
// D# built per CDNA5 ISA.
// ---------------------------------------------------------------------------
__device__ __forceinline__ void tdm_issue_tile(unsigned long long bank_base,
                                               int n, int j0, unsigned lds_addr) {
  unsigned long long ga =
      bank_base + ((unsigned long long)n * BANK1 + (unsigned)j0) * (C_DIM * 4ull);
  int rem_rows = BANK1 - j0;  // OOB rows read as zero (masked in compute)

  v4u g0;
  g0[0] = 1u;                                       // count=1, user mode, no gather
  g0[1] = lds_addr;                                 // LDS byte address of tile
  g0[2] = (unsigned)(ga & 0xFFFFFFFFull);           // global_addr[31:0]
  g0[3] = (unsigned)((ga >> 32) & 0x1FFFFFFull)     // global_addr[56:32]
        | (2u << 30);                               // type = 2 ("image")

  v8i g1;
  g1[0] = (2 << 16)      // data_size = 4 bytes
        | (1 << 20)      // pad_enable
        | (7 << 22)      // pad_interval: every 256 DWORDs (1024B)
        | (3 << 25);     // pad_amount: 4 DWORDs (16B)
  g1[1] = (C_DIM & 0xFFFF) << 16;                   // tensor_dim0[15:0] = 256
  g1[2] = (rem_rows & 0xFFFF) << 16;                // tensor_dim1[15:0]
  g1[3] = (C_DIM & 0xFFFF) << 16;                   // tile_dim0 = 256
  g1[4] = ROWS_PER_BLOCK;                           // tile_dim1 = 64, tile_dim2 = 0
  g1[5] = C_DIM;                                    // tensor_dim0_stride[31:0] = 256
  g1[6] = 0;
  g1[7] = 0;

  v4i g2 = {};   // 2D tensor: groups 2/3 unused (tile_dim2 == 0)
  v4i g3 = {};
  v8i g4 = {};   // extra group in the 6-arg clang-23 builtin: unused, zero

  __builtin_amdgcn_tensor_load_to_lds(g0, g1, g2, g3, g4, 0);
}

// ---------------------------------------------------------------------------
// Kernel A: dot0[z][n] = <bank[n,0,:], feat[n,:]>   (z=0: video, z=1: text)
// ---------------------------------------------------------------------------
__global__ void dot0_kernel(const float* __restrict__ v_feat,
                            const float* __restrict__ t_feat,
                            const float* __restrict__ v_bank,
                            const float* __restrict__ t_bank,
                            float* __restrict__ ws_dot0) {
  const int n = blockIdx.x;
  const int z = blockIdx.y;
  const float* bank = z ? v_bank : t_bank;   // z=0: video uses t_feat_bank
  const float* feat = z ? t_feat : v_feat;
  __shared__ float red[C_DIM];
  const int c = threadIdx.x;
  red[c] = bank[(size_t)n * BANK1 * C_DIM + c] * feat[n * C_DIM + c];
  __syncthreads();
  for (int s = C_DIM / 2; s > 0; s >>= 1) {
    if (c < s) red[c] += red[c + s];
    __syncthreads();
  }
  if (c == 0) ws_dot0[z * N_BATCH + n] = red[0];
}

// ---------------------------------------------------------------------------
// Kernel B: main streaming kernel. grid = (NBX, N, 2).
// Each wave: 16 j-rows, K-loop of 64 x v_wmma_f32_16x16x4_f32.
// ---------------------------------------------------------------------------
__global__ __launch_bounds__(THREADS) void sim_kernel(
    const float* __restrict__ v_feat, const float* __restrict__ t_feat,
    const float* __restrict__ v_bank, const float* __restrict__ t_bank,
    const float* __restrict__ ws_dot0,
    float* __restrict__ ws_sum, int* __restrict__ ws_cnt) {
  extern __shared__ float smem[];
  float* feat_lds = smem + 2 * (TILE_BYTES / 4);

  const int bx = blockIdx.x;
  const int n  = blockIdx.y;
  const int z  = blockIdx.z;
  const float* bank = z ? v_bank : t_bank;
  const float* feat = z ? t_feat : v_feat;

  // Stage the 256-float feature vector (broadcast-read in the K loop).
  for (int c = threadIdx.x; c < C_DIM; c += THREADS)
    feat_lds[c] = feat[n * C_DIM + c];

  const int lane = threadIdx.x & 31;
  const int wid  = threadIdx.x >> 5;
  const int h    = lane >> 4;                // K-pair select (0 or 2)
  const bool isW0 = (threadIdx.x < 32);      // wave 0 drives the TDM

  // LDS byte offset of dynamic shared segment (== static LDS size).
  const unsigned lds_base = (unsigned)__builtin_amdgcn_groupstaticsize();
  const unsigned long long bank_base = (unsigned long long)(uintptr_t)bank;

  const float dot0 = ws_dot0[z * N_BATCH + n];
  const float e0   = __expf(dot0 * T_INV);

  float sum_acc = 0.0f;
  int   cnt_acc = 0;

  if (isW0) tdm_issue_tile(bank_base, n, bx * ROWS_PER_BLOCK, lds_base);

  int buf = 0;
  for (int jb = bx; jb < NTILES; jb += NBX) {
    const int jbn = jb + NBX;
    if (isW0) {
      if (jbn < NTILES) {
        // prefetch next tile into the other buffer, then wait for current
        tdm_issue_tile(bank_base, n, jbn * ROWS_PER_BLOCK,
                       lds_base + (unsigned)((buf ^ 1) * TILE_BYTES));
        __builtin_amdgcn_s_wait_tensorcnt(1);
      } else {
        __builtin_amdgcn_s_wait_tensorcnt(0);
      }
    }
    __syncthreads();   // tile (and feat on first pass) visible to all waves

    const float* tile  = smem + buf * (TILE_BYTES / 4);
    const float* a_row = tile + (wid * 16 + (lane & 15)) * ROW_STRIDE_F;

    v8f acc = {};
#pragma unroll
    for (int k0 = 0; k0 < C_DIM; k0 += 4) {
      // A: 16x4 f32 tile, lane layout per ISA 7.12.2 (K pair 2h..2h+1)
      v2f a = *(const v2f*)(a_row + k0 + 2 * h);
      // B: feat chunk replicated across all 16 N columns -> D cols identical
      v2f b = *(const v2f*)(feat_lds + k0 + 2 * h);
      acc = __builtin_amdgcn_wmma_f32_16x16x4_f32(
          false, a, false, b, (short)0, acc, false, false);
    }

    // D layout: vgpr r, lanes 0-15 -> M=r ; lanes 16-31 -> M=r+8 (all N equal).
    // Lanes with (lane&8)==0 each own one distinct row.
    const int   r     = lane & 7;
    const int   j     = jb * ROWS_PER_BLOCK + wid * 16 + r + 8 * h;
    const bool  valid = ((lane & 8) == 0) && (j < BANK1);
    const float d     = acc[r];
    const float e     = __expf(d * T_INV);
    if (valid) {
      sum_acc += e;
      if (e > e0) cnt_acc++;        // exp-domain compare: matches reference inf semantics
    }
    __syncthreads();                 // all reads done before buffer reuse
    buf ^= 1;
  }

  // Wave reduction (inactive lanes contributed zeros).
  for (int off = 16; off > 0; off >>= 1) {
    sum_acc += __shfl_down(sum_acc, off);
    cnt_acc += __shfl_down(cnt_acc, off);
  }
  if (lane == 0) {
    atomicAdd(&ws_sum[n], sum_acc);                 // ev and et share one sum
    atomicAdd(&ws_cnt[z * N_BATCH + n], cnt_acc);
  }
}

// ---------------------------------------------------------------------------
// Kernel C: fold 128 per-n partials into the 9 output scalars.
// ---------------------------------------------------------------------------
__global__ void finalize_kernel(const float* __restrict__ ws_dot0,
                                const float* __restrict__ ws_sum,
                                const int* __restrict__ ws_cnt,
                                float* __restrict__ out) {
  __shared__ float red[9][N_BATCH];
  const int n = threadIdx.x;   // 128 threads
  const float sum  = ws_sum[n];
  const float sim0 = __expf(ws_dot0[n] * T_INV) + __expf(ws_dot0[N_BATCH + n] * T_INV);
  const float loss = __logf(sim0 / sum);
  const int pv = ws_cnt[n];
  const int pt = ws_cnt[N_BATCH + n];
  float vals[9];
  vals[0] = loss;
  vals[1] = (pv < 1) ? 1.0f : 0.0f;
  vals[2] = (pv >= 1 && pv < 5) ? 1.0f : 0.0f;
  vals[3] = (pv >= 5 && pv < 10) ? 1.0f : 0.0f;
  vals[4] = (float)(pv + 1);
  vals[5] = (pt < 1) ? 1.0f : 0.0f;
  vals[6] = (pt >= 1 && pt < 5) ? 1.0f : 0.0f;
  vals[7] = (pt >= 5 && pt < 10) ? 1.0f : 0.0f;
  vals[8] = (float)(pt + 1);
  for (int m = 0; m < 9; ++m) red[m][n] = vals[m];
  __syncthreads();
  for (int s = N_BATCH / 2; s > 0; s >>= 1) {
    if (n < s)
      for (int m = 0; m < 9; ++m) red[m][n] += red[m][n + s];
    __syncthreads();
  }
  if (n == 0)
    for (int m = 0; m < 9; ++m) out[m] = red[m][0] * (1.0f / (float)N_BATCH);
}

// ---------------------------------------------------------------------------
extern "C" void kernel_launch(void* const* d_in, const int* in_sizes, int n_in,
                              void* d_out, int out_size, void* d_ws, size_t ws_size,
                              hipStream_t stream) {
  const float* v_feat = (const float*)d_in[0];
  const float* t_feat = (const float*)d_in[1];
  const float* v_bank = (const float*)d_in[2];
  const float* t_bank = (const float*)d_in[3];

  // Workspace layout: [0,512)=sum f32[128]; [512,1536)=cnt i32[2][128];
  //                   [1536,2560)=dot0 f32[2][128]
  float* ws_sum  = (float*)d_ws;
  int*   ws_cnt  = (int*)((char*)d_ws + 512);
  float* ws_dot0 = (float*)((char*)d_ws + 1536);

  (void)hipMemsetAsync(d_ws, 0, 1536, stream);  // zero sum + counters (capture-safe)

  dot0_kernel<<<dim3(N_BATCH, 2), C_DIM, 0, stream>>>(v_feat, t_feat, v_bank,
                                                      t_bank, ws_dot0);

  const size_t shmem = 2 * TILE_BYTES + C_DIM * sizeof(float);  // ~131 KB
  sim_kernel<<<dim3(NBX, N_BATCH, 2), THREADS, shmem, stream>>>(
      v_feat, t_feat, v_bank, t_bank, ws_dot0, ws_sum, ws_cnt);

  finalize_kernel<<<1, N_BATCH, 0, stream>>>(ws_dot0, ws_sum, ws_cnt,
                                             (float*)d_out);
}